// ContinuousDiscreteKalmanFilter_54692113547704
// MI455X (gfx1250) — compile-verified
//
#include <hip/hip_runtime.h>
#include <math.h>

#define DZ   128
#define DX   64
#define KMIX 16
#define TT   2048
#define PSZ  (DZ*DZ)   // 16384
#define NT   256       // 8 wave32s

typedef float v2f __attribute__((ext_vector_type(2)));
typedef float v8f __attribute__((ext_vector_type(8)));

// Dormand–Prince-style tableau from the reference
__constant__ float c_RKA[5][5] = {
  { 0.161f, 0.f, 0.f, 0.f, 0.f },
  { -0.008480655492356989f, 0.335480655492357f, 0.f, 0.f, 0.f },
  { 2.8971530571054935f, -6.359448489975075f, 4.3622954328695815f, 0.f, 0.f },
  { 5.325864828439257f, -11.748883564062828f, 7.4955393428898365f, -0.09249506636175525f, 0.f },
  { 5.86145544294642f, -12.92096931784711f, 8.159367898576159f, -0.071584973281401f, -0.028269050394068383f }
};
__constant__ float c_RKB[6] = {
  0.09646076681806523f, 0.01f, 0.4798896504144996f,
  1.379008574103742f, -3.290069515436081f, 2.324710524099774f
};

__device__ __forceinline__ v2f ldfragA(const float* A, int lda, bool tA, int mrow, int ka) {
  v2f a;
  if (tA) { a.x = A[(ka+0)*lda + mrow]; a.y = A[(ka+1)*lda + mrow]; }
  else    { a.x = A[mrow*lda + (ka+0)]; a.y = A[mrow*lda + (ka+1)]; }
  return a;
}
__device__ __forceinline__ v2f ldfragB(const float* B, int ldb, bool tB, int ncol, int ka) {
  v2f b;
  if (tB) { b.x = B[ncol*ldb + (ka+0)]; b.y = B[ncol*ldb + (ka+1)]; }
  else    { b.x = B[(ka+0)*ldb + ncol]; b.y = B[(ka+1)*ldb + ncol]; }
  return b;
}

// Wave-tiled fp32 GEMM using V_WMMA_F32_16X16X4_F32, 2x2 register blocking:
// each wave owns 2x2 adjacent 16x16 tiles -> 4 WMMAs per 4 fragment loads.
// D[m,n] = asign * sum_k a(m,k)*b(k,n) (+ E[m,n] if E != null)
// a(m,k) = tA ? A[k*lda+m] : A[m*lda+k];  b(k,n) = tB ? B[n*ldb+k] : B[k*ldb+n]
// Mt, Nt must be even.
__device__ __forceinline__ void mm_tiles2(
    const float* A, int lda, bool tA,
    const float* B, int ldb, bool tB,
    float* D, int ldd,
    const float* E, float asign,
    int Mt, int Nt, int Kc, int tid)
{
  const int lane   = tid & 31;
  const int wave   = tid >> 5;
  const int nwaves = NT / 32;
  const int r16    = lane & 15;
  const int khalf  = (lane >> 4) << 1;   // A/B frags: lanes 0-15 -> K{0,1}; 16-31 -> K{2,3}
  const int rbase  = (lane >> 4) << 3;   // C/D: lanes 0-15 -> rows r; 16-31 -> rows r+8
  const int Mb = Mt >> 1, Nb = Nt >> 1;
  const int nblk = Mb * Nb;
  for (int blk = wave; blk < nblk; blk += nwaves) {
    const int bm = blk / Nb, bn = blk % Nb;
    const int mrow0 = bm * 32 + r16, mrow1 = mrow0 + 16;
    const int ncol0 = bn * 32 + r16, ncol1 = ncol0 + 16;
    v8f acc00 = {0.f,0.f,0.f,0.f,0.f,0.f,0.f,0.f};
    v8f acc01 = {0.f,0.f,0.f,0.f,0.f,0.f,0.f,0.f};
    v8f acc10 = {0.f,0.f,0.f,0.f,0.f,0.f,0.f,0.f};
    v8f acc11 = {0.f,0.f,0.f,0.f,0.f,0.f,0.f,0.f};
    for (int k0 = 0; k0 < Kc; k0 += 4) {
      const int ka = k0 + khalf;
      const v2f a0 = ldfragA(A, lda, tA, mrow0, ka);
      const v2f a1 = ldfragA(A, lda, tA, mrow1, ka);
      const v2f b0 = ldfragB(B, ldb, tB, ncol0, ka);
      const v2f b1 = ldfragB(B, ldb, tB, ncol1, ka);
      acc00 = __builtin_amdgcn_wmma_f32_16x16x4_f32(false, a0, false, b0, (short)0, acc00, false, false);
      acc01 = __builtin_amdgcn_wmma_f32_16x16x4_f32(false, a0, false, b1, (short)0, acc01, false, false);
      acc10 = __builtin_amdgcn_wmma_f32_16x16x4_f32(false, a1, false, b0, (short)0, acc10, false, false);
      acc11 = __builtin_amdgcn_wmma_f32_16x16x4_f32(false, a1, false, b1, (short)0, acc11, false, false);
    }
#pragma unroll
    for (int r = 0; r < 8; ++r) {
      const int row0 = bm * 32 + rbase + r, row1 = row0 + 16;
      const int col0 = bn * 32 + r16,       col1 = col0 + 16;
      float v00 = asign * acc00[r], v01 = asign * acc01[r];
      float v10 = asign * acc10[r], v11 = asign * acc11[r];
      if (E) {
        v00 += E[row0 * ldd + col0]; v01 += E[row0 * ldd + col1];
        v10 += E[row1 * ldd + col0]; v11 += E[row1 * ldd + col1];
      }
      D[row0 * ldd + col0] = v00; D[row0 * ldd + col1] = v01;
      D[row1 * ldd + col0] = v10; D[row1 * ldd + col1] = v11;
    }
  }
}

// kP = (eA@Pin) + (eA@Pin)^T + diag(eQ)   [P symmetric => eA P + P eA^T]
// kMu = eA@muIn + eb
__device__ void f_eval(const float* eA, const float* Pin, const float* muIn,
                       float* Mtmp, float* kPo, float* kMuo,
                       const float* eQs, const float* ebs, int tid)
{
  mm_tiles2(eA, DZ, false, Pin, DZ, false, Mtmp, DZ, nullptr, 1.f, 8, 8, DZ, tid);
  __syncthreads();
  for (int e = tid; e < PSZ; e += NT) {
    const int i = e >> 7, j = e & 127;
    float v = Mtmp[e] + Mtmp[j * DZ + i];
    if (i == j) v += eQs[i];
    kPo[e] = v;
  }
  if (tid < DZ) {
    float acc = ebs[tid];
    for (int k = 0; k < DZ; ++k) acc += eA[tid * DZ + k] * muIn[k];
    kMuo[tid] = acc;
  }
  __syncthreads();
}

__global__ __launch_bounds__(NT, 1)
void cdkf_kernel(const float* __restrict__ xs, const float* __restrict__ ts,
                 const float* __restrict__ es, const float* __restrict__ mu0,
                 const float* __restrict__ P0, const float* __restrict__ As,
                 const float* __restrict__ bs, const float* __restrict__ Qc,
                 const float* __restrict__ C,  const float* __restrict__ R,
                 float* __restrict__ outMu, float* __restrict__ outP,
                 float* __restrict__ ws)
{
  __shared__ float Ssh[DX * DX];    // 16 KB: innovation covariance / Cholesky
  __shared__ float Xsh[DX * DZ];    // 32 KB: X = S^{-1} (C P)  -> Kg = X^T
  __shared__ float wsh[KMIX];
  __shared__ float ebs[DZ], eQs[DZ];
  __shared__ float muF[DZ], muIn[DZ];
  __shared__ float kMu[6 * DZ];
  __shared__ float yv[DX];

  const int tid = threadIdx.x;

  float* eA   = ws;                 // 128x128
  float* PF   = ws + 1 * PSZ;       // filtered P (carry)
  float* Pin  = ws + 2 * PSZ;       // stage input / predicted P
  float* Mtmp = ws + 3 * PSZ;       // GEMM temp, reused for Pu_raw
  float* kP   = ws + 4 * PSZ;       // 6 consecutive 128x128 stage slopes
  float* CP   = ws + 10 * PSZ;      // 64x128

  // Init carry: (muF, PF) = (mu0, P0); initial update uses (muIn, Pin) = (mu0, P0)
  for (int e = tid; e < PSZ; e += NT) { const float v = P0[e]; PF[e] = v; Pin[e] = v; }
  if (tid < DZ) { muF[tid] = mu0[tid]; muIn[tid] = mu0[tid]; }
  __syncthreads();

  float tprev = ts[0];

  for (int t = 0; t < TT; ++t) {
    if (tid == 0 && t + 1 < TT) {   // warm L2 for next step's inputs
      __builtin_prefetch(es + (size_t)(t + 1) * KMIX, 0, 0);
      __builtin_prefetch(xs + (size_t)(t + 1) * DX, 0, 0);
    }

    if (t > 0) {
      // ---- per-step constants: w, eA, eb, eQ (invariant over RK stages) ----
      if (tid == 0) {
        float ssum = 0.f;
        for (int k = 0; k < KMIX; ++k) ssum += es[(size_t)t * KMIX + k];
        const float inv = 1.0f / ssum;
        for (int k = 0; k < KMIX; ++k) wsh[k] = es[(size_t)t * KMIX + k] * inv;
      }
      __syncthreads();
      for (int e = tid; e < PSZ; e += NT) {
        float acc = 0.f;
        for (int k = 0; k < KMIX; ++k) acc += wsh[k] * As[(size_t)k * PSZ + e];
        eA[e] = acc;
      }
      if (tid < DZ) {
        float accb = 0.f, accq = 0.f;
        for (int k = 0; k < KMIX; ++k) {
          accb += wsh[k] * bs[k * DZ + tid];
          const float q = Qc[k * DZ + tid];
          accq += wsh[k] * q * q;
        }
        ebs[tid] = accb; eQs[tid] = accq;
      }
      __syncthreads();

      const float tcur = ts[t];
      const float h = tcur - tprev;   // ts strictly increasing -> always integrate
      tprev = tcur;

      // ---- RK45: 6 stages, stage 1 input is the carry ----
      f_eval(eA, PF, muF, Mtmp, kP, kMu, eQs, ebs, tid);
      for (int s = 0; s < 5; ++s) {
        for (int e = tid; e < PSZ; e += NT) {
          float acc = PF[e];
          for (int j = 0; j <= s; ++j) acc += h * c_RKA[s][j] * kP[j * PSZ + e];
          Pin[e] = acc;
        }
        if (tid < DZ) {
          float acc = muF[tid];
          for (int j = 0; j <= s; ++j) acc += h * c_RKA[s][j] * kMu[j * DZ + tid];
          muIn[tid] = acc;
        }
        __syncthreads();
        f_eval(eA, Pin, muIn, Mtmp, kP + (s + 1) * PSZ, kMu + (s + 1) * DZ, eQs, ebs, tid);
      }
      // ---- final combination -> prediction in (muIn, Pin) ----
      for (int e = tid; e < PSZ; e += NT) {
        float acc = PF[e];
        for (int j = 0; j < 6; ++j) acc += h * c_RKB[j] * kP[j * PSZ + e];
        Pin[e] = acc;
      }
      if (tid < DZ) {
        float acc = muF[tid];
        for (int j = 0; j < 6; ++j) acc += h * c_RKB[j] * kMu[j * DZ + tid];
        muIn[tid] = acc;
      }
      __syncthreads();
    }

    // =================== KF measurement update ===================
    // CP = C @ Ppred (64x128)
    mm_tiles2(C, DZ, false, Pin, DZ, false, CP, DZ, nullptr, 1.f, 4, 8, DZ, tid);
    __syncthreads();
    // S = CP @ C^T (64x64), then + R + jitter*I
    mm_tiles2(CP, DZ, false, C, DZ, true, Ssh, DX, nullptr, 1.f, 4, 4, DZ, tid);
    __syncthreads();
    for (int e = tid; e < DX * DX; e += NT) {
      const int i = e >> 6, j = e & 63;
      float v = Ssh[e] + R[e];
      if (i == j) v += 1.0e-4f;
      Ssh[e] = v;
    }
    __syncthreads();
    // In-place lower Cholesky of S (block-cooperative)
    for (int j = 0; j < DX; ++j) {
      if (tid == 0) Ssh[j * DX + j] = sqrtf(Ssh[j * DX + j]);
      __syncthreads();
      const float djj = Ssh[j * DX + j];
      for (int i = j + 1 + tid; i < DX; i += NT) Ssh[i * DX + j] /= djj;
      __syncthreads();
      for (int i = j + 1 + tid; i < DX; i += NT) {
        const float lij = Ssh[i * DX + j];
        for (int k = j + 1; k <= i; ++k) Ssh[i * DX + k] -= lij * Ssh[k * DX + j];
      }
      __syncthreads();
    }
    // Solve S X = CP (128 independent RHS columns, one per thread)
    if (tid < DZ) {
      const int c = tid;
      for (int i = 0; i < DX; ++i) {                  // L z = cp
        float z = CP[i * DZ + c];
        for (int j = 0; j < i; ++j) z -= Ssh[i * DX + j] * Xsh[j * DZ + c];
        Xsh[i * DZ + c] = z / Ssh[i * DX + i];
      }
      for (int i = DX - 1; i >= 0; --i) {             // L^T x = z
        float z = Xsh[i * DZ + c];
        for (int j = i + 1; j < DX; ++j) z -= Ssh[j * DX + i] * Xsh[j * DZ + c];
        Xsh[i * DZ + c] = z / Ssh[i * DX + i];
      }
    }
    // y = x_t - C muPred
    if (tid < DX) {
      float acc = xs[(size_t)t * DX + tid];
      for (int k = 0; k < DZ; ++k) acc -= C[tid * DZ + k] * muIn[k];
      yv[tid] = acc;
    }
    __syncthreads();
    // mu_u = muPred + Kg y = muPred + X^T y
    if (tid < DZ) {
      float acc = muIn[tid];
      for (int j = 0; j < DX; ++j) acc += Xsh[j * DZ + tid] * yv[j];
      muF[tid] = acc;
    }
    // Pu_raw = Ppred - X^T CP   (K = 64)
    mm_tiles2(Xsh, DZ, true, CP, DZ, false, Mtmp, DZ, Pin, -1.f, 8, 8, DX, tid);
    __syncthreads();
    // Symmetrize -> carry, write outputs
    for (int e = tid; e < PSZ; e += NT) {
      const int i = e >> 7, j = e & 127;
      const float v = 0.5f * (Mtmp[e] + Mtmp[j * DZ + i]);
      PF[e] = v;
      outP[(size_t)t * PSZ + e] = v;
    }
    if (tid < DZ) outMu[(size_t)t * DZ + tid] = muF[tid];
    __syncthreads();
  }
}

extern "C" void kernel_launch(void* const* d_in, const int* in_sizes, int n_in,
                              void* d_out, int out_size, void* d_ws, size_t ws_size,
                              hipStream_t stream) {
  (void)in_sizes; (void)n_in; (void)out_size; (void)ws_size;
  const float* xs  = (const float*)d_in[0];
  const float* ts  = (const float*)d_in[1];
  const float* es  = (const float*)d_in[2];
  const float* mu0 = (const float*)d_in[3];
  const float* P0  = (const float*)d_in[4];
  const float* As  = (const float*)d_in[5];
  const float* bs  = (const float*)d_in[6];
  const float* Qc  = (const float*)d_in[7];
  const float* C   = (const float*)d_in[8];
  const float* R   = (const float*)d_in[9];
  float* out = (float*)d_out;
  float* ws  = (float*)d_ws;
  // Sequential scan: single workgroup (8 wave32s) carries (mu, P) through all T steps.
  cdkf_kernel<<<1, NT, 0, stream>>>(xs, ts, es, mu0, P0, As, bs, Qc, C, R,
                                    out, out + (size_t)TT * DZ, ws);
}